// GraphCreator_FS_2D_40510131536545
// MI455X (gfx1250) — compile-verified
//
#include <hip/hip_runtime.h>
#include <hip/hip_bf16.h>

typedef __attribute__((ext_vector_type(2))) float v2f;
typedef __attribute__((ext_vector_type(8))) float v8f;
typedef unsigned long long u64;
typedef unsigned int u32;

#define K_NEIGH 30
#define N_PTS   4096
#define HID     32
#define QPB     128      // queries per block (8 waves * 16)
#define K_HALF  15       // neighbors handled per lane after the pair-merge

static __device__ __forceinline__ u64 shfl_xor16_u64(u64 v) {
    u32 lo = (u32)v, hi = (u32)(v >> 32);
    lo = (u32)__shfl_xor((int)lo, 16, 32);
    hi = (u32)__shfl_xor((int)hi, 16, 32);
    return ((u64)hi << 32) | (u64)lo;
}

// gfx1250 hardware tanh (TRANS op). ISA 7.4: one independent op must follow a
// TRANS op before its result is used; embed a v_nop since the compiler cannot
// see through inline asm to insert it.
static __device__ __forceinline__ float htanh(float x) {
    float r;
    asm volatile("v_tanh_f32 %0, %1\n\tv_nop" : "=v"(r) : "v"(x));
    return r;
}

__global__ __launch_bounds__(256) void knn_itp_wmma_kernel(
    const float* __restrict__ u_g,     // (nu, 4096) labels
    const float* __restrict__ ix_g,    // (nu*4096) point x
    const float* __restrict__ iy_g,    // (nu*4096) point y
    const float* __restrict__ qx_g,    // (nu*Q) query x
    const float* __restrict__ qy_g,    // (nu*Q) query y
    const float* __restrict__ W1_g,    // (2,32)
    const float* __restrict__ b1_g,    // (32)
    const float* __restrict__ W2_g,    // (32,1)
    const float* __restrict__ b2_g,    // (1)
    float* __restrict__ out_g,         // (nu*Q)
    int Q)
{
    __shared__ float2 s_pts[N_PTS];                  // 32KB interleaved coords
    __shared__ __align__(16) float s_tile[8][256];   // 8KB: 16x16 tile per wave
    __shared__ float  s_w1x[HID], s_w1y[HID], s_b1[HID], s_w2[HID];
    __shared__ float  s_b2;

    const int s    = blockIdx.y;
    const int tid  = threadIdx.x;
    const int wave = tid >> 5;
    const int lane = tid & 31;
    const int col  = lane & 15;
    const int hi   = lane >> 4;               // 0: lanes 0-15, 1: lanes 16-31

    // ---- cooperative load of this sample's points + MLP params ----
    {
        const float* px = ix_g + s * N_PTS;
        const float* py = iy_g + s * N_PTS;
        for (int i = tid; i < N_PTS; i += 256)
            s_pts[i] = make_float2(px[i], py[i]);
        if (tid < HID) {
            s_w1x[tid] = W1_g[tid];        // W1[0][h]
            s_w1y[tid] = W1_g[HID + tid];  // W1[1][h]
            s_b1[tid]  = b1_g[tid];
            s_w2[tid]  = W2_g[tid];
        }
        if (tid == 0) s_b2 = b2_g[0];
    }
    __syncthreads();

    // ---- lane pair (l, l+16) shares query (l&15) of this wave's 16 ----
    const int qi = blockIdx.x * QPB + wave * 16 + col;
    const float qx = qx_g[s * Q + qi];
    const float qy = qy_g[s * Q + qi];
    const float qn = qx * qx + qy * qy;

    // A-matrix 16x4 f32: lanes 0-15 -> K=0,1 ; lanes 16-31 -> K=2,3
    // row(q) = [qx, qy, |q|^2, 1]
    v2f amat;
    amat.x = hi ? qn   : qx;
    amat.y = hi ? 1.0f : qy;

    // per-lane top-30 as packed keys: (f32 bits of d2)<<32 | point index.
    u64 best[K_NEIGH];
#pragma unroll
    for (int k = 0; k < K_NEIGH; ++k) best[k] = ~0ull;

    float* tile = &s_tile[wave][0];
    const float4* rowp = (const float4*)&s_tile[wave][col * 16 + 8 * hi];

    for (int jt = 0; jt < N_PTS; jt += 16) {
        const float2 p = s_pts[jt + col];
        const float pn = p.x * p.x + p.y * p.y;

        // B-matrix 4x16 f32: lanes 0-15 -> K=0,1 ; lanes 16-31 -> K=2,3
        // col(p) = [-2px, -2py, 1, |p|^2]
        v2f bmat;
        bmat.x = hi ? 1.0f : (-2.0f * p.x);
        bmat.y = hi ? pn   : (-2.0f * p.y);

        v8f c = {0.f, 0.f, 0.f, 0.f, 0.f, 0.f, 0.f, 0.f};
        // 16x16 tile of exact squared distances in a single WMMA
        c = __builtin_amdgcn_wmma_f32_16x16x4_f32(false, amat, false, bmat,
                                                  (short)0, c, false, false);

        // scatter C layout (reg r, lane) -> tile[row][col], row = r + 8*hi
#pragma unroll
        for (int r = 0; r < 8; ++r)
            tile[(r + 8 * hi) * 16 + col] = c[r];
        // same-wave LDS: DS pipe keeps these writes ahead of the reads below

        // each lane grabs its 8 candidates with two b128 loads, then scans
        const float4 r0 = rowp[0];
        const float4 r1 = rowp[1];
        const float cand[8] = {r0.x, r0.y, r0.z, r0.w, r1.x, r1.y, r1.z, r1.w};
#pragma unroll
        for (int cc = 0; cc < 8; ++cc) {
            float v = __builtin_amdgcn_fmed3f(cand[cc], 0.0f, 3.0e38f);
            u64 key = ((u64)__float_as_uint(v) << 32) |
                      (u64)(u32)(jt + 8 * hi + cc);
            if (key < best[K_NEIGH - 1]) {
                u64 cur = key;                 // branchless min/max bubble
#pragma unroll
                for (int k = 0; k < K_NEIGH; ++k) {
                    u64 bk = best[k];
                    bool lt = cur < bk;
                    u64 lo2 = lt ? cur : bk;
                    u64 hi2 = lt ? bk  : cur;
                    best[k] = lo2;
                    cur = hi2;
                }
            }
        }
    }

    // ---- merge the lane pair's two sorted top-30 lists (bitonic min step):
    // 30 smallest of (A, B) = { min(A[k], B[29-k]) }.  Each lane keeps the
    // half it computes -> the pair jointly covers all 30 exactly once.
    u64 sel[K_HALF];
#pragma unroll
    for (int j = K_NEIGH - K_HALF; j < K_NEIGH; ++j) {
        u64 part = shfl_xor16_u64(best[j]);
        u64 mine = best[K_NEIGH - 1 - j];
        sel[K_NEIGH - 1 - j] = (part < mine) ? part : mine;
    }

    // ---- MLP logits for this lane's 15 neighbors ----
    const float* lab_g = u_g + s * N_PTS;
    float lg[K_HALF], lb[K_HALF];
    float mx = -3.0e38f;
#pragma unroll 3
    for (int k = 0; k < K_HALF; ++k) {
        const int j = (int)(u32)sel[k];
        const float rx = s_pts[j].x - qx;
        const float ry = s_pts[j].y - qy;
        float acc = s_b2;
#pragma unroll 8
        for (int h = 0; h < HID; ++h) {
            float t = htanh(fmaf(rx, s_w1x[h], fmaf(ry, s_w1y[h], s_b1[h])));
            acc = fmaf(t, s_w2[h], acc);
        }
        lg[k] = acc;
        lb[k] = lab_g[j];
        mx = fmaxf(mx, acc);
    }

    // ---- pairwise-stable softmax + weighted label sum ----
    float gmax = fmaxf(mx, __shfl_xor(mx, 16, 32));
    float den = 0.0f, num = 0.0f;
#pragma unroll
    for (int k = 0; k < K_HALF; ++k) {
        float e = __expf(lg[k] - gmax);
        den += e;
        num = fmaf(e, lb[k], num);
    }
    den += __shfl_xor(den, 16, 32);
    num += __shfl_xor(num, 16, 32);

    if (!hi)
        out_g[s * Q + qi] = num / den;
}

extern "C" void kernel_launch(void* const* d_in, const int* in_sizes, int n_in,
                              void* d_out, int out_size, void* d_ws, size_t ws_size,
                              hipStream_t stream) {
    const float* u      = (const float*)d_in[0];
    const float* init_x = (const float*)d_in[1];
    const float* init_y = (const float*)d_in[2];
    const float* x      = (const float*)d_in[3];
    const float* y      = (const float*)d_in[4];
    const float* W1     = (const float*)d_in[5];
    const float* b1     = (const float*)d_in[6];
    const float* W2     = (const float*)d_in[7];
    const float* b2     = (const float*)d_in[8];
    float* out          = (float*)d_out;

    const int nu = in_sizes[0] / N_PTS;     // 4
    const int Q  = in_sizes[3] / nu;        // 16384

    dim3 grid(Q / QPB, nu);
    knn_itp_wmma_kernel<<<grid, 256, 0, stream>>>(
        u, init_x, init_y, x, y, W1, b1, W2, b2, out, Q);
}